// VocoMorph_21328807591980
// MI455X (gfx1250) — compile-verified
//
#include <hip/hip_runtime.h>

typedef __attribute__((ext_vector_type(2))) float v2f;
typedef __attribute__((ext_vector_type(8))) float v8f;

#define WMMA_F32X4(a, b, c) \
  __builtin_amdgcn_wmma_f32_16x16x4_f32(false, (a), false, (b), (short)0, (c), false, false)

namespace vc {
constexpr int B_ = 8;
constexpr int T_ = 480000;
constexpr int CHUNK = 4096, STRIDE = 3072;
constexpr int NFFT = 1024, HOP = 256, F = 513, NF = 13;
constexpr int H = 256, E = 128, G3 = 3 * H;      // 768
constexpr int KX = F + E;                        // 641
constexpr int NCH = 157;                         // ceil(T_/STRIDE)
constexpr int NSEQ = B_ * NCH;                   // 1256
constexpr int MR = NSEQ * NF;                    // 16328
constexpr int FP = 528;                          // padded F (multiple of 16)
constexpr int FK = 516;                          // F padded to multiple of 4 (iDFT K)
constexpr int KP = 656;                          // KX padded to multiple of 16 (xg K)
constexpr int TPAD = (NCH - 1) * STRIDE + CHUNK; // 483328
constexpr int PF = 2 * F;                        // 1026

// ---- workspace layout (float offsets); inner GEMM loops are branch-free ----
constexpr size_t OFF_WIN   = 0;                                    // 1024
constexpr size_t OFF_COSBT = 1024;                                 // FP*NFFT  [n][k]
constexpr size_t OFF_SINBT = OFF_COSBT + (size_t)FP * NFFT;        // FP*NFFT  [n][k]
constexpr size_t OFF_DCT   = OFF_SINBT + (size_t)FP * NFFT;        // NFFT*FK  [n][k], k>=F zeroed
constexpr size_t OFF_DST_  = OFF_DCT + (size_t)NFFT * FK;          // NFFT*FK  [n][k]
constexpr size_t OFF_PA    = OFF_DST_ + (size_t)NFFT * FK;         // B_*TPAD zero-padded audio
constexpr size_t OFF_RE    = OFF_PA + (size_t)B_ * TPAD;           // MR*FP (cols>=F zeroed)
constexpr size_t OFF_IM    = OFF_RE + (size_t)MR * FP;             // MR*FP
constexpr size_t OFF_X     = OFF_IM + (size_t)MR * FP;             // MR*KP  [mag|emb|0]
constexpr size_t OFF_XG    = OFF_X + (size_t)MR * KP;              // MR*G3
constexpr size_t OFF_HS    = OFF_XG + (size_t)MR * G3;             // MR*H
constexpr size_t OFF_HG    = OFF_HS + (size_t)MR * H;              // NSEQ*G3
constexpr size_t OFF_WXT   = OFF_HG + (size_t)NSEQ * G3;           // G3*KP   [n][k], k>=KX zeroed
constexpr size_t OFF_WHT   = OFF_WXT + (size_t)G3 * KP;            // G3*H    [n][k]
constexpr size_t OFF_PWT   = OFF_WHT + (size_t)G3 * H;             // PF*H    [n][k]
constexpr size_t OFF_Y     = OFF_XG;  // aliases xg+hs exactly: MR*NFFT == MR*(G3+H)

constexpr int MT_MR   = (MR + 15) / 16;     // 1021
constexpr int MT_SEQ  = (NSEQ + 15) / 16;   // 79
constexpr int NT_F    = (F + 15) / 16;      // 33
constexpr int NT_G3   = G3 / 16;            // 48
constexpr int NT_NFFT = NFFT / 16;          // 64
constexpr float PI2N  = 6.28318530717958647692f / 1024.0f;
}  // namespace vc

using namespace vc;

__device__ __forceinline__ v2f ld2(const float* p) { return *(const v2f*)p; }

// ---------------- init: window, bases, padded audio, transposed weights ----------------
__global__ void k_init_win(float* ws) {
  int t = blockIdx.x * blockDim.x + threadIdx.x;
  if (t < NFFT) ws[OFF_WIN + t] = 0.5f - 0.5f * cosf((float)t * PI2N);
}

__global__ void k_init_fwd_basis(float* ws) {  // [n][k], n<FP, k<NFFT
  int idx = blockIdx.x * blockDim.x + threadIdx.x;
  if (idx >= FP * NFFT) return;
  int n = idx / NFFT, t = idx % NFFT;
  float ang = (float)((n * t) & (NFFT - 1)) * PI2N;
  ws[OFF_COSBT + idx] = cosf(ang);
  ws[OFF_SINBT + idx] = -sinf(ang);  // stored im == true rfft imag part
}

__global__ void k_init_inv_basis(float* ws) {  // [n][k], n<NFFT, k<FK, k>=F -> 0
  int idx = blockIdx.x * blockDim.x + threadIdx.x;
  if (idx >= NFFT * FK) return;
  int n = idx / FK, f = idx % FK;
  if (f >= F) { ws[OFF_DCT + idx] = 0.0f; ws[OFF_DST_ + idx] = 0.0f; return; }
  float ang = (float)((n * f) & (NFFT - 1)) * PI2N;
  bool edge = (f == 0) || (f == NFFT / 2);
  float sc = (edge ? 1.0f : 2.0f) / (float)NFFT;
  ws[OFF_DCT + idx] = sc * cosf(ang);
  ws[OFF_DST_ + idx] = edge ? 0.0f : -sc * sinf(ang);
}

__global__ void k_pad_audio(const float* __restrict__ audio, float* __restrict__ ws) {
  int idx = blockIdx.x * blockDim.x + threadIdx.x;
  if (idx >= B_ * TPAD) return;
  int b = idx / TPAD, t = idx % TPAD;
  ws[OFF_PA + idx] = (t < T_) ? audio[(size_t)b * T_ + t] : 0.0f;
}

// dst[n*KPdst + k] = (k < K) ? src[k*LDsrc + n] : 0   for n in [0,N)
__global__ void k_transpose(float* __restrict__ dst, const float* __restrict__ src,
                            int K, int LDsrc, int N, int KPdst) {
  int idx = blockIdx.x * blockDim.x + threadIdx.x;
  if (idx >= N * KPdst) return;
  int n = idx / KPdst, k = idx % KPdst;
  dst[idx] = (k < K) ? src[(size_t)k * LDsrc + n] : 0.0f;
}

__global__ void k_emb_fill(const float* __restrict__ emb_table,
                           const int* __restrict__ effect_id, float* __restrict__ ws) {
  int idx = blockIdx.x * blockDim.x + threadIdx.x;
  constexpr int W = KP - F;  // 143
  if (idx >= MR * W) return;
  int m = idx / W, j = idx % W;
  int b = m / (NF * NCH);
  float v = (j < E) ? emb_table[(size_t)effect_id[b] * E + j] : 0.0f;
  ws[OFF_X + (size_t)m * KP + F + j] = v;
}

// ---------------- forward DFT as WMMA GEMM (branch-free inner loop) ----------------
__global__ void k_dft(float* __restrict__ ws) {
  const float* win   = ws + OFF_WIN;
  const float* cosBT = ws + OFF_COSBT;
  const float* sinBT = ws + OFF_SINBT;
  const float* pa    = ws + OFF_PA;
  float* re = ws + OFF_RE;
  float* im = ws + OFF_IM;
  float* x  = ws + OFF_X;
  int wid = (blockIdx.x * blockDim.x + threadIdx.x) >> 5;
  int lane = threadIdx.x & 31;
  if (wid >= MT_MR * NT_F) return;
  int mt = wid / NT_F, nt = wid % NT_F;
  int mrow = mt * 16 + (lane & 15);
  int ncol = nt * 16 + (lane & 15);
  int ksub = (lane >> 4) << 1;
  int mm = (mrow < MR) ? mrow : 0;  // row-0 clamp: results for pad rows discarded at store
  int seq = mm / NF, nf = mm % NF;
  int b = seq / NCH, ch = seq % NCH;
  const float* arow = pa + (size_t)b * TPAD + ch * STRIDE + nf * HOP;  // always in-bounds
  const float* bcrow = cosBT + (size_t)ncol * NFFT;
  const float* bsrow = sinBT + (size_t)ncol * NFFT;

  v8f accR = {}, accI = {};
#pragma unroll 4
  for (int k0 = 0; k0 < NFFT; k0 += 4) {
    int kk = k0 + ksub;
    v2f a = ld2(arow + kk) * ld2(win + kk);
    accR = WMMA_F32X4(a, ld2(bcrow + kk), accR);
    accI = WMMA_F32X4(a, ld2(bsrow + kk), accI);
  }
  int rhi = (lane >> 4) * 8;
#pragma unroll
  for (int r = 0; r < 8; ++r) {
    int ms = mt * 16 + rhi + r;
    if (ms < MR) {
      bool v = ncol < F;
      float rr = v ? accR[r] : 0.0f;
      float ii = v ? accI[r] : 0.0f;
      size_t ix = (size_t)ms * FP + ncol;
      re[ix] = rr;               // pad cols 513..527 become 0 (branch-free iDFT)
      im[ix] = ii;
      if (v) x[(size_t)ms * KP + ncol] = sqrtf(rr * rr + ii * ii);  // mag into x
    }
  }
}

// ---------------- xg = [mag|emb] @ gru_wx (x and wxT both zero-padded) ----------------
__global__ void k_xg(float* __restrict__ ws) {
  const float* x   = ws + OFF_X;
  const float* wxT = ws + OFF_WXT;
  float* xg = ws + OFF_XG;
  int wid = (blockIdx.x * blockDim.x + threadIdx.x) >> 5;
  int lane = threadIdx.x & 31;
  if (wid >= MT_MR * NT_G3) return;
  int mt = wid / NT_G3, nt = wid % NT_G3;
  int mrow = mt * 16 + (lane & 15);
  int ncol = nt * 16 + (lane & 15);
  int ksub = (lane >> 4) << 1;
  const float* arow = x + (size_t)((mrow < MR) ? mrow : 0) * KP;
  const float* brow = wxT + (size_t)ncol * KP;

  v8f acc = {};
#pragma unroll 4
  for (int k0 = 0; k0 < KP; k0 += 4) {
    int kk = k0 + ksub;
    acc = WMMA_F32X4(ld2(arow + kk), ld2(brow + kk), acc);
  }
  int rhi = (lane >> 4) * 8;
#pragma unroll
  for (int r = 0; r < 8; ++r) {
    int ms = mt * 16 + rhi + r;
    if (ms < MR) xg[(size_t)ms * G3 + ncol] = acc[r];
  }
}

// ---------------- GRU step: hg = h_{t-1} @ wh + b (t==0 -> bias only) ----------------
__global__ void k_hg(const float* __restrict__ gb, float* __restrict__ ws, int t) {
  const float* hs  = ws + OFF_HS;
  const float* whT = ws + OFF_WHT;
  float* hg = ws + OFF_HG;
  int wid = (blockIdx.x * blockDim.x + threadIdx.x) >> 5;
  int lane = threadIdx.x & 31;
  if (wid >= MT_SEQ * NT_G3) return;
  int mt = wid / NT_G3, nt = wid % NT_G3;
  int mrow = mt * 16 + (lane & 15);
  int ncol = nt * 16 + (lane & 15);
  int ksub = (lane >> 4) << 1;

  v8f acc = {};
  if (t > 0) {  // wave-uniform scalar branch
    const float* arow = hs + (size_t)(((mrow < NSEQ) ? mrow : 0) * NF + (t - 1)) * H;
    const float* brow = whT + (size_t)ncol * H;
#pragma unroll 4
    for (int k0 = 0; k0 < H; k0 += 4) {
      int kk = k0 + ksub;
      acc = WMMA_F32X4(ld2(arow + kk), ld2(brow + kk), acc);
    }
  }
  float bias = gb[ncol];
  int rhi = (lane >> 4) * 8;
#pragma unroll
  for (int r = 0; r < 8; ++r) {
    int ms = mt * 16 + rhi + r;
    if (ms < NSEQ) hg[(size_t)ms * G3 + ncol] = acc[r] + bias;
  }
}

// ---------------- GRU gates ----------------
__global__ void k_gate(float* __restrict__ ws, int t) {
  int idx = blockIdx.x * blockDim.x + threadIdx.x;
  if (idx >= NSEQ * H) return;
  int seq = idx / H, j = idx % H;
  const float* xgr = ws + OFF_XG + (size_t)(seq * NF + t) * G3;
  const float* hgr = ws + OFF_HG + (size_t)seq * G3;
  float* hs = ws + OFF_HS;
  float hprev = (t > 0) ? hs[(size_t)(seq * NF + t - 1) * H + j] : 0.0f;
  float z = 1.0f / (1.0f + expf(-(xgr[j] + hgr[j])));
  float r = 1.0f / (1.0f + expf(-(xgr[H + j] + hgr[H + j])));
  float n = tanhf(xgr[2 * H + j] + r * hgr[2 * H + j]);
  hs[(size_t)(seq * NF + t) * H + j] = (1.0f - z) * n + z * hprev;
}

// ---------------- proj + FiLM (gamma/beta dual-acc share A; fused rescale) ----------------
__global__ void k_proj_film(const float* __restrict__ pb, float* __restrict__ ws) {
  const float* hs  = ws + OFF_HS;
  const float* pwT = ws + OFF_PWT;
  float* re = ws + OFF_RE;
  float* im = ws + OFF_IM;
  int wid = (blockIdx.x * blockDim.x + threadIdx.x) >> 5;
  int lane = threadIdx.x & 31;
  if (wid >= MT_MR * NT_F) return;
  int mt = wid / NT_F, nt = wid % NT_F;
  int mrow = mt * 16 + (lane & 15);
  int ncol = nt * 16 + (lane & 15);
  int ncb = (ncol < F) ? ncol : F - 1;  // clamp keeps B loads in-bounds
  int ksub = (lane >> 4) << 1;
  const float* arow = hs + (size_t)((mrow < MR) ? mrow : 0) * H;
  const float* bgrow = pwT + (size_t)ncb * H;
  const float* bbrow = pwT + (size_t)(F + ncb) * H;

  v8f accG = {}, accB = {};
#pragma unroll 4
  for (int k0 = 0; k0 < H; k0 += 4) {
    int kk = k0 + ksub;
    v2f a = ld2(arow + kk);
    accG = WMMA_F32X4(a, ld2(bgrow + kk), accG);
    accB = WMMA_F32X4(a, ld2(bbrow + kk), accB);
  }
  float bg0 = pb[ncb], bb0 = pb[F + ncb];
  int rhi = (lane >> 4) * 8;
#pragma unroll
  for (int r = 0; r < 8; ++r) {
    int ms = mt * 16 + rhi + r;
    if (ms < MR && ncol < F) {
      size_t ix = (size_t)ms * FP + ncol;
      float rr = re[ix], ii = im[ix];
      float mag = sqrtf(rr * rr + ii * ii);
      float mmod = (accG[r] + bg0) * mag + (accB[r] + bb0);
      if (mag > 1e-20f) {
        float s = mmod / mag;
        re[ix] = rr * s;
        im[ix] = ii * s;
      } else {  // phase = atan2(0,0) = 0 -> cos=1, sin=0
        re[ix] = mmod;
        im[ix] = 0.0f;
      }
    }
  }
}

// ---------------- iDFT as WMMA GEMM (re/im pad cols & dcT/dsT pad rows are 0) ----------------
__global__ void k_idft(float* __restrict__ ws) {
  const float* re  = ws + OFF_RE;
  const float* im  = ws + OFF_IM;
  const float* dcT = ws + OFF_DCT;
  const float* dsT = ws + OFF_DST_;
  const float* win = ws + OFF_WIN;
  float* y = ws + OFF_Y;
  int wid = (blockIdx.x * blockDim.x + threadIdx.x) >> 5;
  int lane = threadIdx.x & 31;
  if (wid >= MT_MR * NT_NFFT) return;
  int mt = wid / NT_NFFT, nt = wid % NT_NFFT;
  int mrow = mt * 16 + (lane & 15);
  int ncol = nt * 16 + (lane & 15);
  int ksub = (lane >> 4) << 1;
  int mm = (mrow < MR) ? mrow : 0;
  const float* rrow = re + (size_t)mm * FP;
  const float* irow = im + (size_t)mm * FP;
  const float* bcrow = dcT + (size_t)ncol * FK;
  const float* bsrow = dsT + (size_t)ncol * FK;

  v8f acc = {};
#pragma unroll 4
  for (int k0 = 0; k0 < FK; k0 += 4) {
    int kk = k0 + ksub;
    acc = WMMA_F32X4(ld2(rrow + kk), ld2(bcrow + kk), acc);
    acc = WMMA_F32X4(ld2(irow + kk), ld2(bsrow + kk), acc);
  }
  float w = win[ncol];
  int rhi = (lane >> 4) * 8;
#pragma unroll
  for (int r = 0; r < 8; ++r) {
    int ms = mt * 16 + rhi + r;
    if (ms < MR) y[(size_t)ms * NFFT + ncol] = acc[r] * w;
  }
}

// ---------------- overlap-add with analytic wsum normalization ----------------
__global__ void k_ola(float* __restrict__ out, const float* __restrict__ ws) {
  int idx = blockIdx.x * blockDim.x + threadIdx.x;
  if (idx >= B_ * T_) return;
  int b = idx / T_, t = idx % T_;
  const float* y = ws + OFF_Y;
  const float* win = ws + OFF_WIN;
  float sum = 0.0f;
  int kmax = t / STRIDE;
  for (int k = kmax; k >= 0 && k >= kmax - 1; --k) {
    int p = t - k * STRIDE;
    if (p >= CHUNK || k >= NCH) continue;
    int nf_hi = p / HOP;
    if (nf_hi > NF - 1) nf_hi = NF - 1;
    int nf_lo = (p > NFFT - 1) ? (p - (NFFT - 1) + HOP - 1) / HOP : 0;
    float rec = 0.0f, wsum = 0.0f;
    for (int nf = nf_lo; nf <= nf_hi; ++nf) {
      int n = p - nf * HOP;
      rec += y[(size_t)((b * NCH + k) * NF + nf) * NFFT + n];
      float w = win[n];
      wsum += w * w;
    }
    sum += rec / fmaxf(wsum, 1e-8f);
  }
  out[idx] = sum;
}

extern "C" void kernel_launch(void* const* d_in, const int* in_sizes, int n_in,
                              void* d_out, int out_size, void* d_ws, size_t ws_size,
                              hipStream_t stream) {
  (void)in_sizes; (void)n_in; (void)out_size; (void)ws_size;
  const float* audio     = (const float*)d_in[0];
  const float* emb_table = (const float*)d_in[1];
  const float* gru_wx    = (const float*)d_in[2];
  const float* gru_wh    = (const float*)d_in[3];
  const float* gru_b     = (const float*)d_in[4];
  const float* proj_w    = (const float*)d_in[5];
  const float* proj_b    = (const float*)d_in[6];
  const int*   effect_id = (const int*)d_in[7];
  float* out = (float*)d_out;
  float* ws  = (float*)d_ws;

  auto blk = [](long long n) { return (unsigned)((n + 255) / 256); };

  k_init_win<<<blk(NFFT), 256, 0, stream>>>(ws);
  k_init_fwd_basis<<<blk((long long)FP * NFFT), 256, 0, stream>>>(ws);
  k_init_inv_basis<<<blk((long long)NFFT * FK), 256, 0, stream>>>(ws);
  k_pad_audio<<<blk((long long)B_ * TPAD), 256, 0, stream>>>(audio, ws);
  k_transpose<<<blk((long long)G3 * KP), 256, 0, stream>>>(ws + OFF_WXT, gru_wx, KX, G3, G3, KP);
  k_transpose<<<blk((long long)G3 * H), 256, 0, stream>>>(ws + OFF_WHT, gru_wh, H, G3, G3, H);
  k_transpose<<<blk((long long)PF * H), 256, 0, stream>>>(ws + OFF_PWT, proj_w, H, PF, PF, H);

  k_dft<<<blk((long long)MT_MR * NT_F * 32), 256, 0, stream>>>(ws);
  k_emb_fill<<<blk((long long)MR * (KP - F)), 256, 0, stream>>>(emb_table, effect_id, ws);
  k_xg<<<blk((long long)MT_MR * NT_G3 * 32), 256, 0, stream>>>(ws);

  for (int t = 0; t < NF; ++t) {
    k_hg<<<blk((long long)MT_SEQ * NT_G3 * 32), 256, 0, stream>>>(gru_b, ws, t);
    k_gate<<<blk((long long)NSEQ * H), 256, 0, stream>>>(ws, t);
  }

  k_proj_film<<<blk((long long)MT_MR * NT_F * 32), 256, 0, stream>>>(proj_b, ws);
  k_idft<<<blk((long long)MT_MR * NT_NFFT * 32), 256, 0, stream>>>(ws);
  k_ola<<<blk((long long)B_ * T_), 256, 0, stream>>>(out, ws);
}